// PSU_LIF_5789615915465
// MI455X (gfx1250) — compile-verified
//
#include <hip/hip_runtime.h>

typedef __attribute__((ext_vector_type(2))) float v2f;
typedef __attribute__((ext_vector_type(8))) float v8f;

namespace {
constexpr int kB = 16;     // batch
constexpr int kT = 1024;   // time steps
constexpr int kH = 1024;   // hidden channels
constexpr int kChunk = 16; // time steps per WMMA tile
constexpr int kNChunks = kT / kChunk;
}

// One wave32 per hidden channel h.
// Chunked linear scan: V_chunk = M_h @ X_chunk + b^(i+1) * carry,
// with M_h[i][j] = b_h^(i-j) (lower triangular), batch = WMMA N dim.
__global__ __launch_bounds__(32)
void lif_pscan_wmma(const float* __restrict__ x,
                    const float* __restrict__ beta,
                    float* __restrict__ spk,
                    float* __restrict__ vout)
{
  const int h    = blockIdx.x;      // hidden channel
  const int lane = threadIdx.x;     // 0..31
  const int col  = lane & 15;       // batch column (N) of B/C/D fragments
  const int half = lane >> 4;       // 0 or 1 (lane half)

  float b = beta[h];
  b = fminf(fmaxf(b, 0.0f), 1.0f);

  // Powers b^0 .. b^16 (b^0 = 1 so b==0 degenerates correctly to V_t = x_t).
  float pw[17];
  pw[0] = 1.0f;
  #pragma unroll
  for (int i = 1; i <= 16; ++i) pw[i] = pw[i - 1] * b;

  // A fragments for the 4 K-slices of the 16x16 triangular matrix M.
  // ISA 16x4 f32 A layout: row = lane&15 for both halves;
  // VGPR0 holds K = 2*half, VGPR1 holds K = 2*half+1 (within the slice).
  v2f A[4];
  #pragma unroll
  for (int j = 0; j < 4; ++j) {
    const int row = col;
    const int c0  = 4 * j + 2 * half;
    const int c1  = c0 + 1;
    A[j].x = (row >= c0) ? pw[row - c0] : 0.0f;
    A[j].y = (row >= c1) ? pw[row - c1] : 0.0f;
  }

  // Carry for batch column `col`, replicated across both lane halves.
  float carryV = 0.0f, carryR = 0.0f;

  const float* xp = x + (size_t)col * kT * kH + h;

  // Preload chunk 0 in B layout (4x16 slice: col = lane&15,
  // VGPR0 row = 4j+2*half, VGPR1 row = 4j+2*half+1).
  v2f Bx[4];
  #pragma unroll
  for (int j = 0; j < 4; ++j) {
    const int r = 4 * j + 2 * half;
    Bx[j].x = xp[(r    ) * kH];
    Bx[j].y = xp[(r + 1) * kH];
  }

  for (int c = 0; c < kNChunks; ++c) {
    const int t0 = c * kChunk;

    // Local membrane scan: V_local = M @ X_chunk  (4 chained K=4 WMMAs).
    v8f vc = {};
    #pragma unroll
    for (int j = 0; j < 4; ++j)
      vc = __builtin_amdgcn_wmma_f32_16x16x4_f32(false, A[j], false, Bx[j],
                                                 (short)0, vc, false, false);

    // Prefetch next chunk's X while the carry chain resolves.
    if (c + 1 < kNChunks) {
      const int tn = t0 + kChunk;
      #pragma unroll
      for (int j = 0; j < 4; ++j) {
        const int r = tn + 4 * j + 2 * half;
        Bx[j].x = xp[(r    ) * kH];
        Bx[j].y = xp[(r + 1) * kH];
      }
    }

    // Carry-in: V[i][n] += b^(i+1) * carryV, row i = k + 8*half.
    #pragma unroll
    for (int k = 0; k < 8; ++k)
      vc[k] += pw[k + 8 * half + 1] * carryV;

    // sigmoid(V) via fast transcendentals (v_exp_f32 + v_rcp_f32) —
    // keeps the sequential carry path off the IEEE division sequence.
    float s[8];
    #pragma unroll
    for (int k = 0; k < 8; ++k)
      s[k] = __builtin_amdgcn_rcpf(1.0f + __expf(-vc[k]));

    // New V carry = V row 15 of this chunk (lane col+16, VGPR 7).
    carryV = __shfl(vc[7], col + 16, 32);

    // Re-stripe sigmoid(V) from D layout into B layout for the R matmul.
    // S[row][n] lives in lane (n + 16*(row>=8)), register s[row&7];
    // columns already match (both are lane&15), only rows move.
    v2f Bs[4];
    #pragma unroll
    for (int j = 0; j < 4; ++j) {
      const int r0 = 4 * j;      // rows needed by lane half 0
      const int r2 = 4 * j + 2;  // rows needed by lane half 1
      float x0 = __shfl(s[(r0    ) & 7], col + (((r0    ) >> 3) << 4), 32);
      float x1 = __shfl(s[(r0 + 1) & 7], col + (((r0 + 1) >> 3) << 4), 32);
      float y0 = __shfl(s[(r2    ) & 7], col + (((r2    ) >> 3) << 4), 32);
      float y1 = __shfl(s[(r2 + 1) & 7], col + (((r2 + 1) >> 3) << 4), 32);
      Bs[j].x = half ? y0 : x0;
      Bs[j].y = half ? y1 : x1;
    }

    // Local refractory scan: R_local = M @ sigmoid(V)_chunk.
    v8f rc = {};
    #pragma unroll
    for (int j = 0; j < 4; ++j)
      rc = __builtin_amdgcn_wmma_f32_16x16x4_f32(false, A[j], false, Bs[j],
                                                 (short)0, rc, false, false);
    #pragma unroll
    for (int k = 0; k < 8; ++k)
      rc[k] += pw[k + 8 * half + 1] * carryR;

    carryR = __shfl(rc[7], col + 16, 32);

    // Emit spike = heaviside(V - R - 1) and V.
    float* vp = vout + (size_t)col * kT * kH + h;
    float* sp = spk  + (size_t)col * kT * kH + h;
    #pragma unroll
    for (int k = 0; k < 8; ++k) {
      const int t = t0 + k + 8 * half;
      const float v = vc[k];
      vp[t * kH] = v;
      sp[t * kH] = (v - rc[k] - 1.0f > 0.0f) ? 1.0f : 0.0f;
    }
  }
}

extern "C" void kernel_launch(void* const* d_in, const int* in_sizes, int n_in,
                              void* d_out, int out_size, void* d_ws, size_t ws_size,
                              hipStream_t stream) {
  const float* x    = (const float*)d_in[0];   // [16, 1024, 1024] f32
  const float* beta = (const float*)d_in[1];   // [1024] f32
  float* spk  = (float*)d_out;                 // output 0: spikes
  float* vout = spk + (size_t)kB * kT * kH;    // output 1: V
  lif_pscan_wmma<<<kH, 32, 0, stream>>>(x, beta, spk, vout);
}